// Parzen_34892314313015
// MI455X (gfx1250) — compile-verified
//
#include <hip/hip_runtime.h>
#include <stdint.h>

typedef __attribute__((ext_vector_type(2))) float v2f;
typedef __attribute__((ext_vector_type(4))) float v4f;
typedef __attribute__((ext_vector_type(8))) float v8f;
typedef __attribute__((ext_vector_type(4))) unsigned int u32x4;
typedef __attribute__((ext_vector_type(8))) int i32x8;
typedef __attribute__((ext_vector_type(4))) int i32x4;

#define NPTS 8192
#define MCTR 16384
#define DDIM 64
#define NSPLIT 4
#define WAVES_PER_BLOCK 8
#define TILES_PER_SPLIT ((MCTR / 16) / NSPLIT)        /* 256 */
#define TILES_PER_STAGE 4                             /* 64 mu rows / stage */
#define STAGES (TILES_PER_SPLIT / TILES_PER_STAGE)    /* 64 */
#define ROW_STRIDE 68     /* 64 data + 4 pad floats: (4L+4c+2h)%64 hits all 64 banks once */
#define STAGE_FLOATS (TILES_PER_STAGE * 16 * ROW_STRIDE) /* 4352 */

__device__ __forceinline__ float xor16(float v) { return __shfl_xor(v, 16, 32); }

// ---------------------------------------------------------------------------
// Prep: q[m] = (0.5/sigma^2) * log2(e) * ||mu_m||^2
// ---------------------------------------------------------------------------
__global__ void parzen_mu2_kernel(const float* __restrict__ mu,
                                  const float* __restrict__ sigma,
                                  float* __restrict__ q) {
  int m = blockIdx.x * blockDim.x + threadIdx.x;
  if (m >= MCTR) return;
  const v4f* row = reinterpret_cast<const v4f*>(mu + m * DDIM);
  float s = 0.f;
#pragma unroll
  for (int i = 0; i < DDIM / 4; ++i) {
    v4f v = row[i];
    s += v[0] * v[0] + v[1] * v[1] + v[2] * v[2] + v[3] * v[3];
  }
  float sg = sigma[0];
  float a = 0.5f / (sg * sg);
  q[m] = (a * 1.4426950408889634f) * s;
}

// ---------------------------------------------------------------------------
// Main: TDM double-buffered mu staging in LDS + f32 WMMA (two independent
// accumulator chains) + streaming base-2 logsumexp epilogue.
// ---------------------------------------------------------------------------
__global__ void __launch_bounds__(WAVES_PER_BLOCK * 32)
parzen_main_kernel(const float* __restrict__ x,
                   const float* __restrict__ mu,
                   const float* __restrict__ sigma,
                   const float* __restrict__ q,
                   float* __restrict__ pmax,
                   float* __restrict__ psum) {
  __shared__ float smu[2 * STAGE_FLOATS];   // 2 x 17408 B, padded rows

  const int lane = threadIdx.x & 31;
  const int wave = threadIdx.x >> 5;
  const int nn = lane & 15;   // x column / mu row within 16-wide tile
  const int h = lane >> 4;    // half-wave selector
  const int x0 = (blockIdx.x * WAVES_PER_BLOCK + wave) * 16;
  const int split = blockIdx.y;
  const int t0 = split * TILES_PER_SPLIT;   // first 16-row mu tile of this split

  const float LOG2E = 1.4426950408889634f;
  const float sg = sigma[0];
  const float a = 0.5f / (sg * sg);

  // Persistent B operand: x tile (16 rows x 64 K) + row sum-of-squares.
  // B 4x16 layout: lane n<16 holds K={4c,4c+1}, lane n+16 holds K={4c+2,4c+3}.
  v2f bx[16];
  float ss = 0.f;
  const float* xrow = x + (x0 + nn) * DDIM + 2 * h;
#pragma unroll
  for (int c = 0; c < 16; ++c) {
    v2f t = *reinterpret_cast<const v2f*>(xrow + 4 * c);
    ss += t[0] * t[0] + t[1] * t[1];
    bx[c] = t;
  }
  ss += xor16(ss);
  const float p2 = (a * LOG2E) * ss;   // uniform per-column shift, applied at end
  const float cs = 2.f * a * LOG2E;    // fold 2a*log2(e) into x registers
#pragma unroll
  for (int c = 0; c < 16; ++c) bx[c] = bx[c] * cs;

  // --- Tensor DMA descriptor (D#) ---------------------------------------
  // Group 1: data_size=4B (2), pad_enable, pad_interval=64 DW (code 5),
  // pad_amount=4 DW (code 3) -> LDS row stride 68 floats. 1-D tile of 4096
  // elements (16 KB contiguous in global memory).
  i32x8 g1;
  g1[0] = (2 << 16) | (1 << 20) | (5 << 22) | (3 << 25);
  g1[1] = (int)((4096u & 0xFFFFu) << 16);    // tensor_dim0 lo16
  g1[2] = (int)((4096u >> 16) | (1u << 16)); // tensor_dim0 hi16 | tensor_dim1=1 lo16
  g1[3] = (int)(4096u << 16);                // tile_dim0 = 4096
  g1[4] = 0;                                 // tile_dim1 = tile_dim2 = 0 (1-D)
  g1[5] = 4096;                              // tensor_dim0_stride lo32
  g1[6] = 0;
  g1[7] = 0;
  // Groups 2/3: identity-size higher dims (dims=1, tile dims 0=unused) so a
  // hardware 5-D interpretation degenerates to the same 1-D copy.
  i32x4 g2;
  g2[0] = 1;                 // tensor_dim2 = 1
  g2[1] = 1;                 // tensor_dim3 = 1 (iterate disabled)
  g2[2] = 0;                 // tensor_dim2_stride lo32 = 0
  g2[3] = 0;                 // stride hi | tile_dim3 = 0 (unused)
  i32x4 g3;
  g3[0] = 0;                 // tensor_dim3_stride lo32
  g3[1] = (int)(1u << 16);   // stride hi16=0 | tensor_dim4=1 lo16
  g3[2] = 0;                 // tensor_dim4 hi16=0 | tile_dim4=0 (unused)
  g3[3] = 0;
  const i32x8 gpad = {0, 0, 0, 0, 0, 0, 0, 0};  // 6-arg form: extra group, zeros
  const unsigned lds_base =
      (unsigned)(unsigned long long)(uintptr_t)&smu[0];  // low 32 bits = LDS offset

  auto tdm_issue = [&](int stage, int buf) {
    unsigned long long ga = (unsigned long long)(uintptr_t)(
        mu + (size_t)(t0 + stage * TILES_PER_STAGE) * 16 * DDIM);
    u32x4 g0;
    g0[0] = 1u;                                            // count=1 descriptor
    g0[1] = lds_base + (unsigned)buf * (STAGE_FLOATS * 4); // lds_addr (bytes)
    g0[2] = (unsigned)ga;                                  // global_addr lo32
    g0[3] = (unsigned)((ga >> 32) & 0x01FFFFFFu) | (2u << 30); // addr hi | type=2
    __builtin_amdgcn_tensor_load_to_lds(g0, g1, g2, g3, gpad, 0);
  };

  if (wave == 0) tdm_issue(0, 0);

  float runmax = -3.402823466e38f;
  float runsum = 0.f;

  for (int s = 0; s < STAGES; ++s) {
    const int cur = s & 1;
    if (wave == 0) __builtin_amdgcn_s_wait_tensorcnt(0);  // stage s landed
    __syncthreads();                                      // visible to all waves
    if (wave == 0 && s + 1 < STAGES) tdm_issue(s + 1, cur ^ 1);  // prefetch

    const float* buf = &smu[cur * STAGE_FLOATS];
#pragma unroll
    for (int j = 0; j < TILES_PER_STAGE; ++j) {
      const int t = t0 + s * TILES_PER_STAGE + j;  // global mu tile index
      // A 16x4 layout: lane m<16 holds K={4c,4c+1}, lane m+16 K={4c+2,4c+3}.
      const float* arow = buf + (j * 16 + nn) * ROW_STRIDE + 2 * h;
      v2f am[16];
#pragma unroll
      for (int c = 0; c < 16; ++c)
        am[c] = *reinterpret_cast<const v2f*>(arow + 4 * c);

      // Two independent accumulator chains for matrix-pipe ILP.
      v8f acc0 = {};
      v8f acc1 = {};
#pragma unroll
      for (int c = 0; c < 8; ++c) {
        acc0 = __builtin_amdgcn_wmma_f32_16x16x4_f32(false, am[2 * c], false,
                                                     bx[2 * c], (short)0, acc0,
                                                     false, false);
        acc1 = __builtin_amdgcn_wmma_f32_16x16x4_f32(false, am[2 * c + 1], false,
                                                     bx[2 * c + 1], (short)0, acc1,
                                                     false, false);
      }

      const v4f* qp = reinterpret_cast<const v4f*>(q + t * 16 + 8 * h);
      v4f q0v = qp[0];
      v4f q1v = qp[1];

      float v0 = (acc0[0] + acc1[0]) - q0v[0];
      float v1 = (acc0[1] + acc1[1]) - q0v[1];
      float v2 = (acc0[2] + acc1[2]) - q0v[2];
      float v3 = (acc0[3] + acc1[3]) - q0v[3];
      float v4 = (acc0[4] + acc1[4]) - q1v[0];
      float v5 = (acc0[5] + acc1[5]) - q1v[1];
      float v6 = (acc0[6] + acc1[6]) - q1v[2];
      float v7 = (acc0[7] + acc1[7]) - q1v[3];

      float tmax = fmaxf(fmaxf(fmaxf(v0, v1), fmaxf(v2, v3)),
                         fmaxf(fmaxf(v4, v5), fmaxf(v6, v7)));
      float nmax = fmaxf(runmax, tmax);
      runsum = runsum * __builtin_exp2f(runmax - nmax)
             + __builtin_exp2f(v0 - nmax) + __builtin_exp2f(v1 - nmax)
             + __builtin_exp2f(v2 - nmax) + __builtin_exp2f(v3 - nmax)
             + __builtin_exp2f(v4 - nmax) + __builtin_exp2f(v5 - nmax)
             + __builtin_exp2f(v6 - nmax) + __builtin_exp2f(v7 - nmax);
      runmax = nmax;
    }
  }

  // Combine the two half-waves (mu rows i and i+8 of every tile).
  float omax = xor16(runmax);
  float osum = xor16(runsum);
  float fm = fmaxf(runmax, omax);
  float fs = runsum * __builtin_exp2f(runmax - fm) +
             osum * __builtin_exp2f(omax - fm);
  if (h == 0) {
    pmax[split * NPTS + x0 + nn] = fm - p2;
    psum[split * NPTS + x0 + nn] = fs;
  }
}

// ---------------------------------------------------------------------------
// Finalize: combine NSPLIT partial (max, sum) pairs per point + constants.
// ---------------------------------------------------------------------------
__global__ void parzen_final_kernel(const float* __restrict__ pmax,
                                    const float* __restrict__ psum,
                                    const float* __restrict__ sigma,
                                    float* __restrict__ out) {
  int n = blockIdx.x * blockDim.x + threadIdx.x;
  if (n >= NPTS) return;
  float mx = -3.402823466e38f;
#pragma unroll
  for (int s = 0; s < NSPLIT; ++s) mx = fmaxf(mx, pmax[s * NPTS + n]);
  float sum = 0.f;
#pragma unroll
  for (int s = 0; s < NSPLIT; ++s)
    sum += psum[s * NPTS + n] * __builtin_exp2f(pmax[s * NPTS + n] - mx);
  const float LN2 = 0.6931471805599453f;
  float sg = sigma[0];
  float lse = LN2 * (mx + __builtin_log2f(sum));
  // -log(16384) = -14*ln2 ; -log(sigma*sqrt(2pi)) = -log(sigma) - 0.918938533...
  out[n] = lse - 14.f * LN2 - (__builtin_logf(sg) + 0.91893853320467274f);
}

// ---------------------------------------------------------------------------
extern "C" void kernel_launch(void* const* d_in, const int* in_sizes, int n_in,
                              void* d_out, int out_size, void* d_ws, size_t ws_size,
                              hipStream_t stream) {
  (void)in_sizes; (void)n_in; (void)out_size; (void)ws_size;
  const float* x = (const float*)d_in[0];
  const float* mu = (const float*)d_in[1];
  const float* sigma = (const float*)d_in[2];
  float* out = (float*)d_out;

  float* q = (float*)d_ws;             // MCTR floats (64 KB)
  float* pmax = q + MCTR;              // NSPLIT*NPTS floats (128 KB)
  float* psum = pmax + NSPLIT * NPTS;  // NSPLIT*NPTS floats (128 KB)

  parzen_mu2_kernel<<<MCTR / 256, 256, 0, stream>>>(mu, sigma, q);

  dim3 grid(NPTS / (16 * WAVES_PER_BLOCK), NSPLIT);  // (64, 4)
  parzen_main_kernel<<<grid, WAVES_PER_BLOCK * 32, 0, stream>>>(x, mu, sigma, q,
                                                                pmax, psum);

  parzen_final_kernel<<<(NPTS + 255) / 256, 256, 0, stream>>>(pmax, psum, sigma, out);
}